// PointWarping_5291399708683
// MI455X (gfx1250) — compile-verified
//
#include <hip/hip_runtime.h>
#include <hip/hip_bf16.h>

typedef float v2f __attribute__((ext_vector_type(2)));
typedef float v8f __attribute__((ext_vector_type(8)));

#define BLOCK_THREADS     256   // 8 waves (wave32)
#define QUERIES_PER_BLOCK 128   // 8 waves * 16 queries each
#define CHUNK_PTS         2048  // candidate points staged in LDS (32 KB as float4)
#define EPSV              1e-10f

// key = |cand|^2 - 2*dot(cand, query); differs from true sqdist by a per-query
// constant (|q|^2), so top-k ordering per query is identical.
__global__ __launch_bounds__(BLOCK_THREADS)
void pw_knn_warp_kernel(const float* __restrict__ xyz1,
                        const float* __restrict__ xyz2,
                        const float* __restrict__ flow1,
                        float* __restrict__ out,
                        int N)   // N1 == N2 == N
{
    __shared__ float4 s_cand[CHUNK_PTS];   // {x, y, z, |p|^2} augmented candidates

    const int tid    = threadIdx.x;
    const int lane   = tid & 31;
    const int wave   = tid >> 5;
    const int lane16 = lane & 15;
    const bool hi    = (lane >= 16);

    const int blocksPerBatch = N / QUERIES_PER_BLOCK;
    const int b     = blockIdx.x / blocksPerBatch;
    const int qbase = (blockIdx.x % blocksPerBatch) * QUERIES_PER_BLOCK + wave * 16;

    const float* x1 = xyz1  + (size_t)b * 3 * N;
    const float* x2 = xyz2  + (size_t)b * 3 * N;
    const float* f1 = flow1 + (size_t)b * 3 * N;

    // This lane's query (both lane halves mirror the same 16-query tile).
    const int   q  = qbase + lane16;
    const float qx = x2[0 * N + q];
    const float qy = x2[1 * N + q];
    const float qz = x2[2 * N + q];

    // B-matrix fragment (4x16, KxN), 2 VGPRs:
    //   lanes 0-15 : V0 = K0 = -2*qx, V1 = K1 = -2*qy   (N = lane)
    //   lanes 16-31: V0 = K2 = -2*qz, V1 = K3 = 1.0     (N = lane-16)
    v2f bmat;
    bmat.x = hi ? (-2.0f * qz) : (-2.0f * qx);
    bmat.y = hi ? 1.0f         : (-2.0f * qy);

    // Per-lane LDS base for A fragments (float index into s_cand viewed flat).
    const float* sc    = reinterpret_cast<const float*>(s_cand);
    const float* abase = sc + (size_t)lane16 * 4 + (hi ? 2 : 0);

    // Running top-3 (ascending) selection keys + global candidate indices.
    float d0 = 3.4e38f, d1 = 3.4e38f, d2 = 3.4e38f;
    int   i0 = 0,       i1 = 0,       i2 = 0;

    for (int cbase = 0; cbase < N; cbase += CHUNK_PTS) {
        // Cooperative load + augmentation of candidate chunk into LDS.
        for (int m = tid; m < CHUNK_PTS; m += BLOCK_THREADS) {
            const int g = cbase + m;
            const float cx = x1[0 * N + g] + f1[0 * N + g];
            const float cy = x1[1 * N + g] + f1[1 * N + g];
            const float cz = x1[2 * N + g] + f1[2 * N + g];
            s_cand[m] = make_float4(cx, cy, cz, cx * cx + cy * cy + cz * cz);
            // Hint: pull the next chunk toward the caches while we compute.
            if (cbase + CHUNK_PTS < N) {
                __builtin_prefetch(&x1[0 * N + g + CHUNK_PTS], 0, 1);
                __builtin_prefetch(&f1[0 * N + g + CHUNK_PTS], 0, 1);
            }
        }
        __syncthreads();

        // Software-pipelined A-fragment: preload tile 0, then each iteration
        // issues the ds_load for tile t+1 before running tile t's selection
        // chain, so s_wait_dscnt hides behind ~100 VALU ops of top-k work.
        float2 av = *reinterpret_cast<const float2*>(abase);

        #pragma unroll 4
        for (int mt = 0; mt < CHUNK_PTS; mt += 16) {
            // A-matrix fragment (16x4, MxK), 2 VGPRs:
            //   lanes 0-15 : K0 = x, K1 = y          (M = lane)
            //   lanes 16-31: K2 = z, K3 = |cand|^2   (M = lane-16)
            v2f amat; amat.x = av.x; amat.y = av.y;

            // Prefetch next tile's fragment (wraps to tile 0 on the final
            // iteration: redundant but valid, keeps the loop branch-free,
            // and still precedes the chunk barrier).
            const int mtn = (mt + 16) & (CHUNK_PTS - 1);
            av = *reinterpret_cast<const float2*>(abase + (size_t)mtn * 4);

            v8f c = {};
            c = __builtin_amdgcn_wmma_f32_16x16x4_f32(
                    false, amat, false, bmat, (short)0, c, false, false);
            // c[v] = |cand_m|^2 - 2*dot(cand_m, query_n)
            //   lanes 0-15 : n = lane,    m = mt + v
            //   lanes 16-31: n = lane-16, m = mt + 8 + v
            const int mg = cbase + mt + (hi ? 8 : 0);
            #pragma unroll
            for (int v = 0; v < 8; ++v) {
                const float d = c[v];
                const int   i = mg + v;
                const bool c2v = d < d2, c1v = d < d1, c0v = d < d0;
                d2 = c1v ? d1 : (c2v ? d : d2);
                i2 = c1v ? i1 : (c2v ? i : i2);
                d1 = c0v ? d0 : (c1v ? d : d1);
                i1 = c0v ? i0 : (c1v ? i : i1);
                d0 = c0v ? d  : d0;
                i0 = c0v ? i  : i0;
            }
        }
        __syncthreads();
    }

    // Merge the two lane halves (low lanes saw m%16 in 0..7, high in 8..15).
    {
        const float e0 = __shfl_xor(d0, 16, 32);
        const float e1 = __shfl_xor(d1, 16, 32);
        const float e2 = __shfl_xor(d2, 16, 32);
        const int   j0 = __shfl_xor(i0, 16, 32);
        const int   j1 = __shfl_xor(i1, 16, 32);
        const int   j2 = __shfl_xor(i2, 16, 32);
        const float ed[3] = {e0, e1, e2};
        const int   ej[3] = {j0, j1, j2};
        #pragma unroll
        for (int k = 0; k < 3; ++k) {
            const float d = ed[k];
            const int   i = ej[k];
            const bool c2v = d < d2, c1v = d < d1, c0v = d < d0;
            d2 = c1v ? d1 : (c2v ? d : d2);
            i2 = c1v ? i1 : (c2v ? i : i2);
            d1 = c0v ? d0 : (c1v ? d : d1);
            i1 = c0v ? i0 : (c1v ? i : i1);
            d0 = c0v ? d  : d0;
            i0 = c0v ? i  : i0;
        }
    }

    // Epilogue (lanes 0-15 own one query each): exact distances -> IDW weights.
    if (!hi) {
        const int idx[3] = {i0, i1, i2};
        float wk[3], fx[3], fy[3], fz[3];
        float wsum = 0.0f;
        #pragma unroll
        for (int k = 0; k < 3; ++k) {
            const int m = idx[k];
            const float gx = x1[0 * N + m] + f1[0 * N + m];
            const float gy = x1[1 * N + m] + f1[1 * N + m];
            const float gz = x1[2 * N + m] + f1[2 * N + m];
            fx[k] = f1[0 * N + m];
            fy[k] = f1[1 * N + m];
            fz[k] = f1[2 * N + m];
            const float dx = gx - qx, dy = gy - qy, dz = gz - qz;
            float dist = sqrtf(dx * dx + dy * dy + dz * dz);
            dist = fmaxf(dist, EPSV);
            const float inv = 1.0f / dist;
            wk[k] = inv;
            wsum += inv;
        }
        const float rs = 1.0f / wsum;
        float f2x = 0.0f, f2y = 0.0f, f2z = 0.0f;
        #pragma unroll
        for (int k = 0; k < 3; ++k) {
            const float w = wk[k] * rs;
            f2x += w * fx[k];
            f2y += w * fy[k];
            f2z += w * fz[k];
        }
        float* o = out + (size_t)b * 3 * N;
        o[0 * N + q] = qx - f2x;
        o[1 * N + q] = qy - f2y;
        o[2 * N + q] = qz - f2z;
    }
}

extern "C" void kernel_launch(void* const* d_in, const int* in_sizes, int n_in,
                              void* d_out, int out_size, void* d_ws, size_t ws_size,
                              hipStream_t stream) {
    (void)n_in; (void)out_size; (void)d_ws; (void)ws_size;
    const float* xyz1  = (const float*)d_in[0];
    const float* xyz2  = (const float*)d_in[1];
    const float* flow1 = (const float*)d_in[2];
    float* out = (float*)d_out;

    const int B = 2, C = 3;
    const int N = in_sizes[0] / (B * C);   // 8192

    dim3 grid(B * (N / QUERIES_PER_BLOCK));  // 128 blocks
    dim3 block(BLOCK_THREADS);               // 8 waves
    pw_knn_warp_kernel<<<grid, block, 0, stream>>>(xyz1, xyz2, flow1, out, N);
}